// ReLUZ_20899310863245
// MI455X (gfx1250) — compile-verified
//
#include <hip/hip_runtime.h>

// ---- problem constants (match reference: K=1025, F=8192) -------------------
#define KROWS 1025
#define FDIM  8192
#define FDIV4 (FDIM / 4)          // 2048 float4 per row
#define KSPLIT 8
#define ROWS_PER_SPLIT ((KROWS - 1) / KSPLIT)   // 128 error-term rows per split

typedef __attribute__((ext_vector_type(2))) float v2f;
typedef __attribute__((ext_vector_type(4))) float f4;
typedef __attribute__((ext_vector_type(8))) float v8f;

// ---------------------------------------------------------------------------
// Kernel A: abssum partials via V_WMMA_F32_16X16X4_F32 with B = ones.
// Each wave owns 16 features (f0..f0+15) and 128 of the 1024 error rows.
// A-matrix layout (ISA 7.12.2, 32-bit A 16x4): lanes 0-15 -> M=0..15 with
// VGPR0=K0,VGPR1=K1 ; lanes 16-31 -> same M, VGPR0=K2,VGPR1=K3.
// D = A x ones + C  accumulates 4 rows of |x| per WMMA into the f32 C tile.
// D is replicated over N, so lane 0 (N=0) holds abssum for m=0..7 in c[0..7],
// lane 16 holds m=8..15.
// ---------------------------------------------------------------------------
__global__ void zono_abssum_wmma(const float* __restrict__ x,
                                 float* __restrict__ partial) {
  const int wave = (int)((blockIdx.x * blockDim.x + threadIdx.x) >> 5);
  const int lane = threadIdx.x & 31;
  const int ft = wave >> 3;                  // feature tile 0..511
  const int ks = wave & (KSPLIT - 1);        // k-split    0..7
  const int f0 = ft << 4;
  const int row0 = 1 + ks * ROWS_PER_SPLIT;  // skip center row 0
  const int m = lane & 15;
  const int koff = (lane >> 4) << 1;         // 0 for lanes 0-15, 2 for 16-31

  v2f bones; bones.x = 1.0f; bones.y = 1.0f;
  v8f acc = {};

  const float* base = x + (size_t)(row0 + koff) * FDIM + f0 + m;
  #pragma unroll 4
  for (int i = 0; i < ROWS_PER_SPLIT / 4; ++i) {
    const float* p = base + (size_t)i * 4 * FDIM;
    v2f a;
    a.x = __builtin_fabsf(p[0]);        // K = koff
    a.y = __builtin_fabsf(p[FDIM]);     // K = koff + 1
    acc = __builtin_amdgcn_wmma_f32_16x16x4_f32(
        /*neg_a=*/false, a, /*neg_b=*/false, bones,
        /*c_mod=*/(short)0, acc, /*reuse_a=*/false, /*reuse_b=*/false);
  }

  float* outp = partial + (size_t)ks * FDIM + f0;
  if (lane == 0) {
    #pragma unroll
    for (int r = 0; r < 8; ++r) outp[r] = acc[r];        // features f0+0..7
  } else if (lane == 16) {
    #pragma unroll
    for (int r = 0; r < 8; ++r) outp[8 + r] = acc[r];    // features f0+8..15
  }
}

// ---------------------------------------------------------------------------
// Kernel B: fold KSPLIT partials, compute per-feature scale[] and half[].
// ---------------------------------------------------------------------------
__global__ void zono_params(const float* __restrict__ x,
                            const float* __restrict__ lambdas,
                            const float* __restrict__ partial,
                            float* __restrict__ scale,
                            float* __restrict__ halfv) {
  const int f = blockIdx.x * blockDim.x + threadIdx.x;
  if (f >= FDIM) return;
  float s = 0.0f;
  #pragma unroll
  for (int k = 0; k < KSPLIT; ++k) s += partial[k * FDIM + f];
  const float x0  = x[f];                 // row 0 = center
  const float lam = lambdas[f];
  const float l = x0 - s;
  const float u = x0 + s;
  const float pos   = (l > 0.0f) ? 1.0f : 0.0f;
  const float cross = ((u > 0.0f) && (l < 0.0f)) ? 1.0f : 0.0f;
  const float d = fmaxf(-l * lam, u * (1.0f - lam));
  scale[f] = pos + cross * lam;
  halfv[f] = 0.5f * cross * d;
}

// ---------------------------------------------------------------------------
// Kernel C: out[k,f] = scale[f]*x[k,f] (+half[f] on row 0). float4, NT stores.
// x is L2-resident after kernel A; scale/half are tiny and hot.
// ---------------------------------------------------------------------------
__global__ void zono_scale_rows(const float* __restrict__ x,
                                const float* __restrict__ scale,
                                const float* __restrict__ halfv,
                                float* __restrict__ out) {
  const size_t idx = (size_t)blockIdx.x * blockDim.x + threadIdx.x;
  const size_t total = (size_t)KROWS * FDIV4;
  if (idx >= total) return;
  const int c4 = (int)(idx & (FDIV4 - 1));
  const size_t k = idx >> 11;             // FDIV4 == 2048 == 1<<11
  f4 xv = ((const f4*)x)[idx];
  f4 sv = ((const f4*)scale)[c4];
  f4 ov = xv * sv;
  if (k == 0) ov += ((const f4*)halfv)[c4];
  __builtin_nontemporal_store(ov, ((f4*)out) + idx);
}

// ---------------------------------------------------------------------------
// Kernel D: ext block = zeros with diag = half. 268 MB of pure streaming
// stores -> 128-bit non-temporal stores; diagonal touched by 1/2048 threads.
// ---------------------------------------------------------------------------
__global__ void zono_ext_fill(const float* __restrict__ halfv,
                              float* __restrict__ ext) {
  const size_t idx = (size_t)blockIdx.x * blockDim.x + threadIdx.x; // float4 id
  const int c4  = (int)(idx & (FDIV4 - 1));
  const int row = (int)(idx >> 11);
  f4 v = {0.0f, 0.0f, 0.0f, 0.0f};
  if (c4 == (row >> 2)) {                 // this float4 crosses the diagonal
    v[row & 3] = halfv[row];
  }
  __builtin_nontemporal_store(v, ((f4*)ext) + idx);
}

// ---------------------------------------------------------------------------
extern "C" void kernel_launch(void* const* d_in, const int* in_sizes, int n_in,
                              void* d_out, int out_size, void* d_ws, size_t ws_size,
                              hipStream_t stream) {
  (void)in_sizes; (void)n_in; (void)out_size; (void)ws_size;
  const float* x       = (const float*)d_in[0];   // [1025, 8192] f32
  const float* lambdas = (const float*)d_in[1];   // [1, 8192]    f32
  float* out = (float*)d_out;                     // [1025+8192, 8192] f32

  // workspace layout: partial[8][8192] | scale[8192] | half[8192]  (~328 KB)
  float* partial = (float*)d_ws;
  float* scale   = partial + (size_t)KSPLIT * FDIM;
  float* halfv   = scale + FDIM;

  // A: 512 feature tiles x 8 k-splits = 4096 waves = 512 blocks of 256
  zono_abssum_wmma<<<(FDIM / 16) * KSPLIT / 8, 256, 0, stream>>>(x, partial);

  // B: 8192 features
  zono_params<<<FDIM / 256, 256, 0, stream>>>(x, lambdas, partial, scale, halfv);

  // C: 1025*2048 float4 elements
  const size_t totalC = (size_t)KROWS * FDIV4;
  zono_scale_rows<<<(unsigned)((totalC + 255) / 256), 256, 0, stream>>>(
      x, scale, halfv, out);

  // D: 8192*2048 float4 elements into the ext region
  float* ext = out + (size_t)KROWS * FDIM;
  zono_ext_fill<<<(unsigned)(((size_t)FDIM * FDIV4) / 256), 256, 0, stream>>>(
      halfv, ext);
}